// BlinnPhongShaderEnvMap_27041114096024
// MI455X (gfx1250) — compile-verified
//
#include <hip/hip_runtime.h>
#include <math.h>

typedef __attribute__((ext_vector_type(2))) float v2f;
typedef __attribute__((ext_vector_type(8))) float v8f;

#define HW        65536   // H*W pixels (N=1, 256x256)
#define J_LIGHTS  256
#define V_CNT     20000
#define F_CNT     40000
#define EPS_N     1e-6f

// workspace layout in floats
#define WS_VN_OFF  0        // 60000 floats: accumulated vertex normals
#define WS_PIX_OFF 65536    // 65536*8 floats: per-pixel {nx,ny,nz,vx,vy,vz,ndotv,pad}

__global__ void zero_vn_kernel(float* ws) {
    int i = blockIdx.x * blockDim.x + threadIdx.x;
    if (i < V_CNT * 3) ws[WS_VN_OFF + i] = 0.0f;
}

__global__ void face_normals_kernel(const float* __restrict__ verts,
                                    const int* __restrict__ faces,
                                    float* ws) {
    int f = blockIdx.x * blockDim.x + threadIdx.x;
    if (f >= F_CNT) return;
    int i0 = faces[f * 3 + 0];
    int i1 = faces[f * 3 + 1];
    int i2 = faces[f * 3 + 2];
    float ax = verts[i0 * 3 + 0], ay = verts[i0 * 3 + 1], az = verts[i0 * 3 + 2];
    float bx = verts[i1 * 3 + 0], by = verts[i1 * 3 + 1], bz = verts[i1 * 3 + 2];
    float cx = verts[i2 * 3 + 0], cy = verts[i2 * 3 + 1], cz = verts[i2 * 3 + 2];
    float e1x = bx - ax, e1y = by - ay, e1z = bz - az;
    float e2x = cx - ax, e2y = cy - ay, e2z = cz - az;
    float nx = e1y * e2z - e1z * e2y;
    float ny = e1z * e2x - e1x * e2z;
    float nz = e1x * e2y - e1y * e2x;
    float* vn = ws + WS_VN_OFF;
    atomicAdd(&vn[i0 * 3 + 0], nx); atomicAdd(&vn[i0 * 3 + 1], ny); atomicAdd(&vn[i0 * 3 + 2], nz);
    atomicAdd(&vn[i1 * 3 + 0], nx); atomicAdd(&vn[i1 * 3 + 1], ny); atomicAdd(&vn[i1 * 3 + 2], nz);
    atomicAdd(&vn[i2 * 3 + 0], nx); atomicAdd(&vn[i2 * 3 + 1], ny); atomicAdd(&vn[i2 * 3 + 2], nz);
}

__global__ void pixel_prep_kernel(const float* __restrict__ verts,
                                  const int* __restrict__ faces,
                                  const int* __restrict__ p2f,
                                  const float* __restrict__ bary,
                                  const float* __restrict__ cam,
                                  float* ws, float* out) {
    int p = blockIdx.x * blockDim.x + threadIdx.x;
    if (p >= HW) return;
    int f = p2f[p];
    bool mask = (f < 0);
    int fi = mask ? 0 : f;
    float w0 = bary[p * 3 + 0], w1 = bary[p * 3 + 1], w2 = bary[p * 3 + 2];
    int i0 = faces[fi * 3 + 0], i1 = faces[fi * 3 + 1], i2 = faces[fi * 3 + 2];
    const float* vn = ws + WS_VN_OFF;

    float px = 0.f, py = 0.f, pz = 0.f, nx = 0.f, ny = 0.f, nz = 0.f;
    if (!mask) {
        px = w0 * verts[i0 * 3 + 0] + w1 * verts[i1 * 3 + 0] + w2 * verts[i2 * 3 + 0];
        py = w0 * verts[i0 * 3 + 1] + w1 * verts[i1 * 3 + 1] + w2 * verts[i2 * 3 + 1];
        pz = w0 * verts[i0 * 3 + 2] + w1 * verts[i1 * 3 + 2] + w2 * verts[i2 * 3 + 2];
        nx = w0 * vn[i0 * 3 + 0] + w1 * vn[i1 * 3 + 0] + w2 * vn[i2 * 3 + 0];
        ny = w0 * vn[i0 * 3 + 1] + w1 * vn[i1 * 3 + 1] + w2 * vn[i2 * 3 + 1];
        nz = w0 * vn[i0 * 3 + 2] + w1 * vn[i1 * 3 + 2] + w2 * vn[i2 * 3 + 2];
    }
    // normalize pixel normal (x / max(|x|, eps))
    float nlen = sqrtf(nx * nx + ny * ny + nz * nz);
    float ninv = 1.0f / fmaxf(nlen, EPS_N);
    nx *= ninv; ny *= ninv; nz *= ninv;
    // view = normalize(cam - pos)
    float vx = cam[0] - px, vy = cam[1] - py, vz = cam[2] - pz;
    float vlen = sqrtf(vx * vx + vy * vy + vz * vz);
    float vinv = 1.0f / fmaxf(vlen, EPS_N);
    vx *= vinv; vy *= vinv; vz *= vinv;
    float ndotv = nx * vx + ny * vy + nz * vz;

    float* pd = ws + WS_PIX_OFF + (size_t)p * 8;
    pd[0] = nx; pd[1] = ny; pd[2] = nz;
    pd[3] = vx; pd[4] = vy; pd[5] = vz;
    pd[6] = ndotv; pd[7] = 0.f;

    // pixel_normals output: identical for both batches (broadcast then normalize)
    float* nout = out + (size_t)2 * HW * 3;   // after colors
    nout[(size_t)p * 3 + 0] = nx;
    nout[(size_t)p * 3 + 1] = ny;
    nout[(size_t)p * 3 + 2] = nz;
    nout[(size_t)(HW + p) * 3 + 0] = nx;
    nout[(size_t)(HW + p) * 3 + 1] = ny;
    nout[(size_t)(HW + p) * 3 + 2] = nz;
}

// One wave (32 lanes) shades a tile of 16 pixels x 256 lights using
// V_WMMA_F32_16X16X4_F32: 2 WMMAs per 16-light group for n.L / v.L, then 4
// WMMAs accumulating the (shade x light_color) contraction. 8 waves / block.
__global__ void __launch_bounds__(256) shade_kernel(
    const float* __restrict__ ldir,    // (B,J,3)
    const float* __restrict__ lcol,    // (B,J,3)
    const float* __restrict__ shin_p,
    const float* __restrict__ kd_p,
    const float* __restrict__ ks_p,
    const float* __restrict__ ws,
    float* __restrict__ out) {
    __shared__ float s_ldir[J_LIGHTS * 3];
    __shared__ float s_lcol[J_LIGHTS * 3];
    __shared__ float s_ndv[8][16];
    __shared__ float s_tile[8][16][17];

    int tid  = threadIdx.x;
    int wave = tid >> 5;
    int lane = tid & 31;
    int b    = blockIdx.x >> 9;              // 512 blocks per batch
    int tile = blockIdx.x * 8 + wave;        // global 16-pixel tile id
    int tin  = tile & 4095;                  // tile within batch
    int pbase = tin * 16;

    // preload this batch's lights into LDS (256 threads, 1 light each)
    {
        const float* ld = ldir + ((size_t)b * J_LIGHTS + tid) * 3;
        const float* lc = lcol + ((size_t)b * J_LIGHTS + tid) * 3;
        s_ldir[tid * 3 + 0] = ld[0]; s_ldir[tid * 3 + 1] = ld[1]; s_ldir[tid * 3 + 2] = ld[2];
        s_lcol[tid * 3 + 0] = lc[0]; s_lcol[tid * 3 + 1] = lc[1]; s_lcol[tid * 3 + 2] = lc[2];
    }

    int  m16 = lane & 15;
    bool hi  = lane >= 16;

    // A-fragments for this wave's 16 pixels (16x4 layout: v0=K0|K2, v1=K1|K3)
    const float* pd = ws + WS_PIX_OFF + (size_t)(pbase + m16) * 8;
    float nx = pd[0], ny = pd[1], nz = pd[2];
    float vx = pd[3], vy = pd[4], vz = pd[5];
    v2f a_n = { hi ? nz : nx, hi ? 0.0f : ny };
    v2f a_v = { hi ? vz : vx, hi ? 0.0f : vy };
    if (!hi) s_ndv[wave][m16] = pd[6];
    __syncthreads();

    float shin = shin_p[0];
    float kd   = kd_p[0];
    float bpks = ks_p[0] * (shin + 2.0f) / (4.0f * (2.0f - __expf(-shin * 0.5f)));

    v8f cacc = { 0, 0, 0, 0, 0, 0, 0, 0 };
    v8f zero = { 0, 0, 0, 0, 0, 0, 0, 0 };

#pragma unroll 2
    for (int g = 0; g < 16; ++g) {
        // B-fragment: 4x16 light-direction block (rows = xyz0, cols = 16 lights)
        int ln = g * 16 + m16;
        float lx = s_ldir[ln * 3 + 0], ly = s_ldir[ln * 3 + 1], lz = s_ldir[ln * 3 + 2];
        v2f b_l = { hi ? lz : lx, hi ? 0.0f : ly };

        v8f dn = __builtin_amdgcn_wmma_f32_16x16x4_f32(false, a_n, false, b_l,
                                                       (short)0, zero, false, false);
        v8f dv = __builtin_amdgcn_wmma_f32_16x16x4_f32(false, a_v, false, b_l,
                                                       (short)0, zero, false, false);
        // elementwise shade: kd*clip(n.L) + bp*ks*clip((n.v+n.L)*rsq(2+2 v.L))^shin
#pragma unroll
        for (int r = 0; r < 8; ++r) {
            int   m    = r + (hi ? 8 : 0);
            float ndl  = dn[r];
            float vdl  = dv[r];
            float diff = __builtin_amdgcn_fmed3f(ndl, 0.0f, 1.0f);
            // 1/max(sqrt(2+2vdl), eps) == min(rsq(max(2+2vdl,0)), 1/eps)
            float rin  = fminf(__builtin_amdgcn_rsqf(fmaxf(2.0f + 2.0f * vdl, 0.0f)),
                               1.0f / EPS_N);
            float sp   = __builtin_amdgcn_fmed3f((s_ndv[wave][m] + ndl) * rin, 0.0f, 1.0f);
            // sp^shin = exp2(shin*log2(sp)); v_log_f32(0) = -inf -> v_exp_f32 -> 0
            float spp  = __builtin_amdgcn_exp2f(shin * __builtin_amdgcn_logf(sp));
            s_tile[wave][m][m16] = fmaf(kd, diff, bpks * spp);
        }
        // accumulate colors: 4 k-groups of 4 lights against light colors
#pragma unroll
        for (int kg = 0; kg < 4; ++kg) {
            v2f a_s = { s_tile[wave][m16][kg * 4 + (hi ? 2 : 0)],
                        s_tile[wave][m16][kg * 4 + (hi ? 3 : 1)] };
            int l0 = g * 16 + kg * 4 + (hi ? 2 : 0);
            int l1 = g * 16 + kg * 4 + (hi ? 3 : 1);
            v2f b_c = { (m16 < 3) ? s_lcol[l0 * 3 + m16] : 0.0f,
                        (m16 < 3) ? s_lcol[l1 * 3 + m16] : 0.0f };
            cacc = __builtin_amdgcn_wmma_f32_16x16x4_f32(false, a_s, false, b_c,
                                                         (short)0, cacc, false, false);
        }
    }

    // write colors: lane m16 = channel (only 0..2 valid); rows r -> pixels
    if (m16 < 3) {
        float* cout = out + (size_t)b * HW * 3;
#pragma unroll
        for (int r = 0; r < 8; ++r) {
            int m = r + (hi ? 8 : 0);
            cout[(size_t)(pbase + m) * 3 + m16] = cacc[r];
        }
    }
}

extern "C" void kernel_launch(void* const* d_in, const int* in_sizes, int n_in,
                              void* d_out, int out_size, void* d_ws, size_t ws_size,
                              hipStream_t stream) {
    const float* verts = (const float*)d_in[0];
    const int*   faces = (const int*)d_in[1];
    const int*   p2f   = (const int*)d_in[2];
    const float* bary  = (const float*)d_in[3];
    const float* ldir  = (const float*)d_in[4];
    const float* lcol  = (const float*)d_in[5];
    const float* cam   = (const float*)d_in[6];
    const float* shin  = (const float*)d_in[7];
    const float* kd    = (const float*)d_in[8];
    const float* ks    = (const float*)d_in[9];
    float* out = (float*)d_out;
    float* ws  = (float*)d_ws;

    zero_vn_kernel<<<(V_CNT * 3 + 255) / 256, 256, 0, stream>>>(ws);
    face_normals_kernel<<<(F_CNT + 255) / 256, 256, 0, stream>>>(verts, faces, ws);
    pixel_prep_kernel<<<(HW + 255) / 256, 256, 0, stream>>>(verts, faces, p2f, bary, cam, ws, out);
    // 2 batches * 4096 tiles = 8192 waves; 8 waves per 256-thread block
    shade_kernel<<<1024, 256, 0, stream>>>(ldir, lcol, shin, kd, ks, ws, out);
}